// _LoKrLayer_12403865551210
// MI455X (gfx1250) — compile-verified
//
#include <hip/hip_runtime.h>

typedef __attribute__((ext_vector_type(2))) float v2f;
typedef __attribute__((ext_vector_type(4))) float v4f;
typedef __attribute__((ext_vector_type(8))) float v8f;
typedef __attribute__((ext_vector_type(4))) int   v4i;

#define TT        16          // t values per block
#define MROWS     128         // t8 rows per block (TT*8)
#define KC        32          // K-chunk of x staged in LDS
#define XS_STRIDE 36          // 32 + 4 pad dwords: 16B aligned rows, conflict-free
#define US_STRIDE 66          // 64 + 2 pad dwords: 8B aligned rows

// gfx1250 async copy path: global -> LDS without VGPR round-trip (ASYNCcnt).
#if defined(__AMDGCN__) && \
    __has_builtin(__builtin_amdgcn_global_load_async_to_lds_b128) && \
    __has_builtin(__builtin_amdgcn_s_wait_asynccnt)
#define USE_ASYNC_COPY 1
typedef __attribute__((address_space(1))) v4i gv4i;   // global int32x4
typedef __attribute__((address_space(3))) v4i lv4i;   // LDS int32x4
#else
#define USE_ASYNC_COPY 0
#endif

__global__ __launch_bounds__(256) void lokr_fused_kernel(
    const float* __restrict__ x,    // [8192, 4096] == X2 [65536, 512]
    const float* __restrict__ w1,   // [8, 8]
    const float* __restrict__ a,    // [512, 64]
    const float* __restrict__ b,    // [64, 512]
    float* __restrict__ out)        // [8192, 4096]
{
    __shared__ float xs[MROWS * XS_STRIDE];
    __shared__ float us[MROWS * US_STRIDE];

    const int tid  = threadIdx.x;
    const int wave = tid >> 5;
    const int lane = tid & 31;
    const int hf   = lane >> 4;    // lane-half: selects K sub-pair / M+8
    const int l16  = lane & 15;
    const long t8_base = (long)blockIdx.x * MROWS;

    // Warm L2 for the small shared factor matrices (emits global_prefetch_b8).
#pragma unroll
    for (int i = 0; i < 4; ++i) {
        __builtin_prefetch(b + (tid + i * 256) * 32, 0, 3);
        __builtin_prefetch(a + (tid + i * 256) * 32, 0, 3);
    }

    // w1 pre-scaled by alpha/rank = 1/64; uniform address + constant index -> SGPRs
    float w1s[64];
#pragma unroll
    for (int i = 0; i < 64; ++i) w1s[i] = w1[i] * (1.0f / 64.0f);

    // ---------------- GEMM1: z[128 x 64] = X2_tile[128 x 512] @ b^T ----------------
    v8f acc[4];
#pragma unroll
    for (int n = 0; n < 4; ++n) acc[n] = (v8f){0.f,0.f,0.f,0.f,0.f,0.f,0.f,0.f};

    for (int c = 0; c < 512 / KC; ++c) {
        // stage x chunk: 128 rows x 32 cols = 1024 float4, 256 threads x 4
#if USE_ASYNC_COPY
#pragma unroll
        for (int it = 0; it < 4; ++it) {
            int q    = it * 256 + tid;   // float4 index
            int row  = q >> 3;           // 8 float4 per row
            int col4 = q & 7;
            __builtin_amdgcn_global_load_async_to_lds_b128(
                (gv4i*)(x + (t8_base + row) * 512 + c * KC + col4 * 4),
                (lv4i*)(xs + row * XS_STRIDE + col4 * 4),
                /*offset=*/0, /*cpol=*/0);
        }
        __builtin_amdgcn_s_wait_asynccnt(0);
#else
#pragma unroll
        for (int it = 0; it < 4; ++it) {
            int q    = it * 256 + tid;
            int row  = q >> 3;
            int col4 = q & 7;
            v4f v = *(const v4f*)(x + (t8_base + row) * 512 + c * KC + col4 * 4);
            *(v4f*)(xs + row * XS_STRIDE + col4 * 4) = v;
        }
#endif
        __syncthreads();

#pragma unroll
        for (int kk = 0; kk < KC / 4; ++kk) {
            const int il = c * KC + kk * 4;
            // A frag: lane holds M = 16*wave + l16; K pair = kk*4 + hf*2 (+0,+1)
            v2f A = *(const v2f*)(xs + (wave * 16 + l16) * XS_STRIDE + kk * 4 + hf * 2);
#pragma unroll
            for (int n = 0; n < 4; ++n) {
                // B frag: lane holds N = r = n*16 + l16; K pair = il + hf*2
                v2f B = *(const v2f*)(b + (n * 16 + l16) * 512 + il + hf * 2);
                acc[n] = __builtin_amdgcn_wmma_f32_16x16x4_f32(
                    false, A, false, B, (short)0, acc[n], false, false);
            }
        }
        __syncthreads();
    }

    // ---------------- w1 mix: u.v[oi] = sum_ij w1s[oi][ij] * z.v[ij] ----------------
    // C layout: VGPR v -> M = v + 8*hf, so ij = M % 8 = v for both halves.
    v8f u[4];
#pragma unroll
    for (int n = 0; n < 4; ++n) {
#pragma unroll
        for (int oi = 0; oi < 8; ++oi) {
            float s = 0.f;
#pragma unroll
            for (int j = 0; j < 8; ++j) s += w1s[oi * 8 + j] * acc[n][j];
            u[n][oi] = s;
        }
    }

    // stage u to LDS (C layout -> memory), then reload in A layout
#pragma unroll
    for (int v = 0; v < 8; ++v) {
        int row = wave * 16 + v + hf * 8;   // wave-private rows
#pragma unroll
        for (int n = 0; n < 4; ++n)
            us[row * US_STRIDE + n * 16 + l16] = u[n][v];
    }
    __syncthreads();

    // A2 frags for all 16 K-steps (K = r = 64): lane M = 16*wave + l16
    v2f A2[16];
#pragma unroll
    for (int k = 0; k < 16; ++k)
        A2[k] = *(const v2f*)(us + (wave * 16 + l16) * US_STRIDE + k * 4 + hf * 2);

    // ---------------- GEMM2: out_tile[128 x 512] = u[128 x 64] @ (a/64)^T ----------
    // (scale already folded into w1s)
    const long out_base = (long)blockIdx.x * TT * 4096;
    for (int nt = 0; nt < 32; ++nt) {
        v8f d = (v8f){0.f,0.f,0.f,0.f,0.f,0.f,0.f,0.f};
#pragma unroll
        for (int k = 0; k < 16; ++k) {
            // B2 frag: lane N = ok = nt*16 + l16; K pair = k*4 + hf*2
            v2f B2 = *(const v2f*)(a + (nt * 16 + l16) * 64 + k * 4 + hf * 2);
            d = __builtin_amdgcn_wmma_f32_16x16x4_f32(
                false, A2[k], false, B2, (short)0, d, false, false);
        }
        // store: M = v + 8*hf -> t_local = 2*wave + hf, oi = v
        float* op = out + out_base + (long)(2 * wave + hf) * 4096 + nt * 16 + l16;
#pragma unroll
        for (int v = 0; v < 8; ++v) op[v * 512] = d[v];
    }
}

extern "C" void kernel_launch(void* const* d_in, const int* in_sizes, int n_in,
                              void* d_out, int out_size, void* d_ws, size_t ws_size,
                              hipStream_t stream) {
    const float* x  = (const float*)d_in[0];   // [4,2048,4096] f32
    const float* w1 = (const float*)d_in[1];   // [8,8] f32
    const float* a  = (const float*)d_in[2];   // [512,64] f32
    const float* b  = (const float*)d_in[3];   // [64,512] f32
    float* out = (float*)d_out;                // [4,2048,4096] f32

    const int rows   = in_sizes[0] / 4096;     // 8192 t-rows
    const int blocks = rows / TT;              // 512
    hipLaunchKernelGGL(lokr_fused_kernel, dim3(blocks), dim3(256), 0, stream,
                       x, w1, a, b, out);
}